// DyRepHawkesRe_22136261444453
// MI455X (gfx1250) — compile-verified
//
#include <hip/hip_runtime.h>
#include <hip/hip_bf16.h>
#include <math.h>

#define N_NODES 2000
#define HDIM    128
#define BB      300
#define NNEG    10
#define MAXNBR  512
#define NPAIR   (2 * NNEG + 1)   // 21: 1 positive + 20 negative pairs

typedef __attribute__((ext_vector_type(2))) float v2f;
typedef __attribute__((ext_vector_type(8))) float v8f;

// ---------------------------------------------------------------------------
// Kernel 0: copy z0 -> mutable z workspace
// ---------------------------------------------------------------------------
__global__ void dyrep_copy_kernel(const float* __restrict__ src,
                                  float* __restrict__ dst, int n) {
  int i = blockIdx.x * blockDim.x + threadIdx.x;
  if (i < n) dst[i] = src[i];
}

// ---------------------------------------------------------------------------
// Kernel 1: Whz = z0 @ Wh_w^T + Wh_b  via full-precision f32 WMMA 16x16x4.
// One wave per 16x16 output tile: 125 row tiles x 8 col tiles = 1000 waves,
// K=128 -> 32 WMMA issues per wave (compiler software-pipelines the loads).
// ---------------------------------------------------------------------------
__global__ __launch_bounds__(32) void dyrep_whz_init_kernel(
    const float* __restrict__ z0, const float* __restrict__ Wh_w,
    const float* __restrict__ Wh_b, float* __restrict__ whz) {
  const int tile = blockIdx.x;      // 0..999
  const int mt = tile >> 3;         // 0..124
  const int nt = tile & 7;          // 0..7
  const int lane = threadIdx.x;
  const int half = lane >> 4;       // 0: K{0,1}, 1: K{2,3}
  const int l16  = lane & 15;
  const int mBase = mt * 16, nBase = nt * 16;

  v8f c = {};
  for (int k0 = 0; k0 < HDIM; k0 += 4) {
    const int kb = k0 + half * 2;
    v2f a, b;
    // A = z0 tile (16x4), row m = l16
    a.x = z0[(size_t)(mBase + l16) * HDIM + kb];
    a.y = z0[(size_t)(mBase + l16) * HDIM + kb + 1];
    // B = Wh_w^T tile (4x16): B[k][n] = Wh_w[n][k], col n = l16
    b.x = Wh_w[(size_t)(nBase + l16) * HDIM + kb];
    b.y = Wh_w[(size_t)(nBase + l16) * HDIM + kb + 1];
    c = __builtin_amdgcn_wmma_f32_16x16x4_f32(
        /*neg_a=*/false, a, /*neg_b=*/false, b,
        /*c_mod=*/(short)0, c, /*reuse_a=*/false, /*reuse_b=*/false);
  }
  const float bias = Wh_b[nBase + l16];
  // C/D layout: VGPR r -> M = r + half*8, N = l16
  for (int r = 0; r < 8; ++r) {
    const int M = r + half * 8;
    whz[(size_t)(mBase + M) * HDIM + nBase + l16] = c[r] + bias;
  }
}

// ---------------------------------------------------------------------------
// Kernel 2: the sequential 300-step scan. Single workgroup (256 threads =
// 8 wave32). Maintains z and Whz incrementally (only the 2 updated rows per
// step), emits the 21 Hawkes pre-activation dot products g0/g1 per step.
// Latency hiding: float4 (b128) row scans, both nodes concurrently, and
// global_prefetch of the NEXT step's A/S rows overlapping compute phases.
// ---------------------------------------------------------------------------
__global__ __launch_bounds__(256) void dyrep_scan_kernel(
    const int* __restrict__ u, const int* __restrict__ v,
    const float* __restrict__ time_diff, const int* __restrict__ neg,
    const float* __restrict__ A, const float* __restrict__ S,
    const float* __restrict__ Wh_w, const float* __restrict__ Wh_b,
    const float* __restrict__ Wstruct_w, const float* __restrict__ Wstruct_b,
    const float* __restrict__ Wrec_w, const float* __restrict__ Wrec_b,
    const float* __restrict__ Wt_w, const float* __restrict__ Wt_b,
    const float* __restrict__ om0_w, const float* __restrict__ om0_b,
    const float* __restrict__ om1_w, const float* __restrict__ om1_b,
    float* __restrict__ zws, float* __restrict__ whz,
    float* __restrict__ gbuf) {
  __shared__ int   nbr[2][MAXNBR];
  __shared__ float qv[2][MAXNBR];
  __shared__ int   cnt[2];
  __shared__ float qsum[2];
  __shared__ float red[256];
  __shared__ __align__(16) float hs[2][HDIM];
  __shared__ __align__(16) float upd[2][HDIM];

  const int tid = threadIdx.x;
  const int s128 = tid >> 7;     // half-block id: 0 -> node sv, 1 -> node su
  const int j128 = tid & 127;

  for (int step = 0; step < BB; ++step) {
    const int su = u[step];
    const int sv = v[step];

    // ---- Prefetch next step's A/S rows (overlaps with this step's phases;
    //      lowers to global_prefetch_b8). 64 lanes/row x 32-float stride
    //      (128B) covers a 2000-float row.
    if (step + 1 < BB) {
      const int nu = u[step + 1];
      const int nv = v[step + 1];
      const int r = tid >> 6;          // 0..3
      const int o = (tid & 63) * 32;   // float offset, 128B stride
      if (o < N_NODES) {
        const float* row =
            (r == 0) ? &A[(size_t)nu * N_NODES] :
            (r == 1) ? &A[(size_t)nv * N_NODES] :
            (r == 2) ? &S[(size_t)nu * N_NODES] :
                       &S[(size_t)nv * N_NODES];
        __builtin_prefetch(row + o, 0, 3);
      }
    }

    if (tid < 2) cnt[tid] = 0;
    __syncthreads();

    // ---- Phase A1: both nodes concurrently (128 threads each), float4 scan.
    //      s=0 -> agg(sv) = h_struct[0]; s=1 -> agg(su) = h_struct[1].
    {
      const int node = (s128 == 0) ? sv : su;
      const float4* A4 = (const float4*)&A[(size_t)node * N_NODES];
      const float4* S4 = (const float4*)&S[(size_t)node * N_NODES];
      float local = 0.f;
      for (int i = j128; i < N_NODES / 4; i += 128) {
        const float4 a4 = A4[i];
        const float4 sq = S4[i];
        if (a4.x > 0.f) { const float e = expf(sq.x);
          const int idx = atomicAdd(&cnt[s128], 1);
          if (idx < MAXNBR) { nbr[s128][idx] = i * 4 + 0; qv[s128][idx] = e; }
          local += e; }
        if (a4.y > 0.f) { const float e = expf(sq.y);
          const int idx = atomicAdd(&cnt[s128], 1);
          if (idx < MAXNBR) { nbr[s128][idx] = i * 4 + 1; qv[s128][idx] = e; }
          local += e; }
        if (a4.z > 0.f) { const float e = expf(sq.z);
          const int idx = atomicAdd(&cnt[s128], 1);
          if (idx < MAXNBR) { nbr[s128][idx] = i * 4 + 2; qv[s128][idx] = e; }
          local += e; }
        if (a4.w > 0.f) { const float e = expf(sq.w);
          const int idx = atomicAdd(&cnt[s128], 1);
          if (idx < MAXNBR) { nbr[s128][idx] = i * 4 + 3; qv[s128][idx] = e; }
          local += e; }
      }
      red[tid] = local;
      __syncthreads();
      for (int off = 64; off > 0; off >>= 1) {
        if (j128 < off) red[tid] += red[tid + off];
        __syncthreads();
      }
      if (j128 == 0) qsum[s128] = red[tid];
      __syncthreads();
    }

    // ---- Phase A2: h[s][j] = max over neighbors of sigmoid(q_n * Whz[n,j])
    {
      const int c = min(cnt[s128], MAXNBR);
      const float inv = 1.f / (qsum[s128] + 1e-7f);
      float h;
      if (c == 0) {
        h = 0.f;
      } else {
        h = -INFINITY;
        for (int k = 0; k < c; ++k) {
          const int n = nbr[s128][k];
          const float q = qv[s128][k] * inv;
          const float x = q * whz[(size_t)n * HDIM + j128];
          const float sg = 1.f / (1.f + expf(-x));
          h = fmaxf(h, sg);
        }
      }
      hs[s128][j128] = h;
    }
    __syncthreads();

    // ---- Phase B: upd[s] = sigmoid(Wstruct@hs[s] + Wrec@z[row_s] + Wt@td + b)
    {
      const int row = (s128 == 0) ? su : sv;   // z_uv = z[[u, v]]
      float acc = Wstruct_b[j128] + Wrec_b[j128] + Wt_b[j128];
      const float4* hs4 = (const float4*)hs[s128];
      const float4* zr4 = (const float4*)&zws[(size_t)row * HDIM];
      const float4* ws4 = (const float4*)&Wstruct_w[(size_t)j128 * HDIM];
      const float4* wr4 = (const float4*)&Wrec_w[(size_t)j128 * HDIM];
      for (int m = 0; m < HDIM / 4; ++m) {
        const float4 hv = hs4[m], zv = zr4[m], wv = ws4[m], rv = wr4[m];
        acc += wv.x * hv.x + wv.y * hv.y + wv.z * hv.z + wv.w * hv.w;
        acc += rv.x * zv.x + rv.y * zv.y + rv.z * zv.z + rv.w * zv.w;
      }
      const float sdv[4] = {50.f, 7.f, 15.f, 15.f};
      for (int m = 0; m < 4; ++m)
        acc += Wt_w[j128 * 4 + m] *
               (time_diff[(step * 2 + s128) * 4 + m] / sdv[m]);
      upd[s128][j128] = 1.f / (1.f + expf(-acc));
    }
    __syncthreads();

    // ---- Phase C: Hawkes pre-activations on PRE-update z.
    // pair 0 = (z[u], z[v]); pairs 1..20: k<NN -> (z[u], z[neg_k]),
    // k>=NN -> (z[neg_k], z[v]). One wave per pair, float4 per lane.
    {
      const int w = tid >> 5;    // wave id 0..7
      const int lane = tid & 31;
      const float4 o0a = *(const float4*)&om0_w[lane * 4];
      const float4 o0b = *(const float4*)&om0_w[HDIM + lane * 4];
      const float4 o1a = *(const float4*)&om1_w[lane * 4];
      const float4 o1b = *(const float4*)&om1_w[HDIM + lane * 4];
      for (int p = w; p < NPAIR; p += 8) {
        int ru, rv;
        if (p == 0) {
          ru = su; rv = sv;
        } else {
          const int k = p - 1;
          const int nk = neg[step * 2 * NNEG + k];
          ru = (k < NNEG) ? su : nk;
          rv = (k < NNEG) ? nk : sv;
        }
        const float4 zu4 = *(const float4*)&zws[(size_t)ru * HDIM + lane * 4];
        const float4 zv4 = *(const float4*)&zws[(size_t)rv * HDIM + lane * 4];
        float p0 = zu4.x * o0a.x + zu4.y * o0a.y + zu4.z * o0a.z +
                   zu4.w * o0a.w + zv4.x * o0b.x + zv4.y * o0b.y +
                   zv4.z * o0b.z + zv4.w * o0b.w;
        float p1 = zu4.x * o1a.x + zu4.y * o1a.y + zu4.z * o1a.z +
                   zu4.w * o1a.w + zv4.x * o1b.x + zv4.y * o1b.y +
                   zv4.z * o1b.z + zv4.w * o1b.w;
        for (int off = 16; off > 0; off >>= 1) {
          p0 += __shfl_down(p0, off, 32);
          p1 += __shfl_down(p1, off, 32);
        }
        if (lane == 0) {
          gbuf[(step * NPAIR + p) * 2 + 0] = p0 + om0_b[0];
          gbuf[(step * NPAIR + p) * 2 + 1] = p1 + om1_b[0];
        }
      }
    }
    __syncthreads();

    // ---- Phase D: commit z rows, then refresh only the 2 touched Whz rows.
    {
      const int row = (s128 == 0) ? su : sv;
      zws[(size_t)row * HDIM + j128] = upd[s128][j128];
    }
    __syncthreads();
    {
      const int row = (s128 == 0) ? su : sv;
      float acc = Wh_b[j128];
      const float4* zr4 = (const float4*)upd[s128];  // fresh z row (LDS)
      const float4* wj4 = (const float4*)&Wh_w[(size_t)j128 * HDIM];
      for (int m = 0; m < HDIM / 4; ++m) {
        const float4 zv = zr4[m], wv = wj4[m];
        acc += wv.x * zv.x + wv.y * zv.y + wv.z * zv.z + wv.w * zv.w;
      }
      whz[(size_t)row * HDIM + j128] = acc;
    }
    __syncthreads();
  }
}

// ---------------------------------------------------------------------------
// Kernel 3: finalize — apply Hawkes nonlinearity (softplus w/ clip) and write
// lam_uv [B], lam_neg [B,2NN], z_final [N,H] into d_out.
// ---------------------------------------------------------------------------
__global__ void dyrep_finalize_kernel(
    const int* __restrict__ u, const int* __restrict__ v,
    const int* __restrict__ et, const float* __restrict__ t_bar,
    const float* __restrict__ t, const int* __restrict__ neg,
    const float* __restrict__ w_t, const float* __restrict__ alpha,
    const float* __restrict__ psi, const float* __restrict__ gbuf,
    const float* __restrict__ zws, float* __restrict__ out) {
  const int idx = blockIdx.x * blockDim.x + threadIdx.x;
  const int lamCount = BB * NPAIR;
  if (idx < lamCount) {
    const int i = idx / NPAIR;
    const int p = idx % NPAIR;
    const int e = et[i];
    float ts;
    if (p == 0) {
      ts = t[i] - fmaxf(t_bar[(size_t)i * N_NODES + u[i]],
                        t_bar[(size_t)i * N_NODES + v[i]]);
    } else {
      const int k = p - 1;
      const int nk = neg[i * 2 * NNEG + k];
      const int iu = (k < NNEG) ? u[i] : nk;
      const int iv = (k < NNEG) ? nk : v[i];
      ts = t[i] - fmaxf(t_bar[(size_t)i * N_NODES + iu],
                        t_bar[(size_t)i * N_NODES + iv]);
    }
    float g = gbuf[(i * NPAIR + p) * 2 + e];
    g += alpha[e] * expf(-w_t[e] * (ts / 100.f));
    const float ps = psi[e];
    float x = g / (ps + 1e-7f);
    x = fminf(fmaxf(x, -75.f), 75.f);
    const float lam = ps * log1pf(expf(x));
    if (p == 0) out[i] = lam;
    else out[BB + i * (2 * NNEG) + (p - 1)] = lam;
  }
  const int zidx = idx - lamCount;
  if (zidx >= 0 && zidx < N_NODES * HDIM) {
    out[BB + BB * 2 * NNEG + zidx] = zws[zidx];
  }
}

// ---------------------------------------------------------------------------
extern "C" void kernel_launch(void* const* d_in, const int* in_sizes, int n_in,
                              void* d_out, int out_size, void* d_ws,
                              size_t ws_size, hipStream_t stream) {
  (void)in_sizes; (void)n_in; (void)out_size; (void)ws_size;
  const int*   u         = (const int*)  d_in[0];
  const int*   v         = (const int*)  d_in[1];
  const float* time_diff = (const float*)d_in[2];
  const int*   et        = (const int*)  d_in[3];
  const float* t_bar     = (const float*)d_in[4];
  const float* t         = (const float*)d_in[5];
  const int*   neg       = (const int*)  d_in[6];
  const float* z0        = (const float*)d_in[7];
  const float* A         = (const float*)d_in[8];
  const float* S         = (const float*)d_in[9];
  const float* w_t       = (const float*)d_in[10];
  const float* alpha     = (const float*)d_in[11];
  const float* psi       = (const float*)d_in[12];
  const float* om0_w     = (const float*)d_in[13];
  const float* om0_b     = (const float*)d_in[14];
  const float* om1_w     = (const float*)d_in[15];
  const float* om1_b     = (const float*)d_in[16];
  const float* Wh_w      = (const float*)d_in[17];
  const float* Wh_b      = (const float*)d_in[18];
  const float* Wstruct_w = (const float*)d_in[19];
  const float* Wstruct_b = (const float*)d_in[20];
  const float* Wrec_w    = (const float*)d_in[21];
  const float* Wrec_b    = (const float*)d_in[22];
  const float* Wt_w      = (const float*)d_in[23];
  const float* Wt_b      = (const float*)d_in[24];

  float* ws   = (float*)d_ws;
  float* zws  = ws;                          // N*H
  float* whz  = zws + N_NODES * HDIM;        // N*H
  float* gbuf = whz + N_NODES * HDIM;        // B*21*2

  float* out = (float*)d_out;

  // z0 -> mutable z
  {
    const int n = N_NODES * HDIM;
    dyrep_copy_kernel<<<(n + 255) / 256, 256, 0, stream>>>(z0, zws, n);
  }
  // Whz = z0 @ Wh_w^T + Wh_b   (f32 WMMA; 125x8 = 1000 tiles)
  dyrep_whz_init_kernel<<<(N_NODES / 16) * (HDIM / 16), 32, 0, stream>>>(
      z0, Wh_w, Wh_b, whz);
  // sequential scan (single persistent workgroup)
  dyrep_scan_kernel<<<1, 256, 0, stream>>>(
      u, v, time_diff, neg, A, S, Wh_w, Wh_b, Wstruct_w, Wstruct_b, Wrec_w,
      Wrec_b, Wt_w, Wt_b, om0_w, om0_b, om1_w, om1_b, zws, whz, gbuf);
  // finalize lambdas + write z_final
  {
    const int total = BB * NPAIR + N_NODES * HDIM;
    dyrep_finalize_kernel<<<(total + 255) / 256, 256, 0, stream>>>(
        u, v, et, t_bar, t, neg, w_t, alpha, psi, gbuf, zws, out);
  }
}